// GCN_84567906058949
// MI455X (gfx1250) — compile-verified
//
#include <hip/hip_runtime.h>
#include <math.h>

// ---------------------------------------------------------------------------
// GCN forward for MI455X (gfx1250, wave32).
//   h1   = x @ W1                      (WMMA f32 16x16x4, W1 in LDS)
//   agg1 = scatter-add(h1[row]*norm)   (wave/edge, float4 + f32 atomics)
//   h    = relu(agg1 + b1)
//   agg2 = scatter-add(h[row]*norm)
//   pooledF[g] = sum_{n in g} agg2[n]  (atomics into [16,128])
//   out  = log_softmax((pooledF @ W2 + cnt*b2)/max(cnt,1))
// W2 is folded past the scatter+pool (linear ops commute with segment_sum),
// so the 128x2 GEMM runs once on [16,128] instead of per-node.
// ---------------------------------------------------------------------------

#define NF 128          // NFEAT == NHID
#define NG 16           // N_GRAPHS
#define NC 2            // NCLASS

typedef __attribute__((ext_vector_type(2))) float v2f;
typedef __attribute__((ext_vector_type(8))) float v8f;

__device__ __forceinline__ void atomAddF(float* p, float v) {
  unsafeAtomicAdd(p, v);   // lowers to global_atomic_add_f32 on gfx1250
}

// ---------------- degree / normalization --------------------------------
__global__ void deg_kernel(const int* __restrict__ col, float* __restrict__ deg, int E) {
  int e = blockIdx.x * blockDim.x + threadIdx.x;
  if (e < E) atomAddF(&deg[col[e]], 1.0f);
}

__global__ void dinv_kernel(float* __restrict__ deg, int N) {
  int i = blockIdx.x * blockDim.x + threadIdx.x;
  if (i < N) deg[i] = rsqrtf(deg[i] + 1.0f);   // +1 self-loop; deg>=1 always
}

__global__ void norm_kernel(const int* __restrict__ row, const int* __restrict__ col,
                            const float* __restrict__ dinv, float* __restrict__ norm,
                            int E, int Etot) {
  int e = blockIdx.x * blockDim.x + threadIdx.x;
  if (e >= Etot) return;
  int r = (e < E) ? row[e] : (e - E);
  int c = (e < E) ? col[e] : (e - E);
  norm[e] = dinv[r] * dinv[c];
}

// ---------------- dense GEMM: h = x @ W (N x 128 x 128), WMMA f32 --------
// Block: 256 threads = 8 waves. Wave w computes rows [m0, m0+16) x all 128 cols.
__global__ __launch_bounds__(256) void gemm_wmma_f32(const float* __restrict__ x,
                                                     const float* __restrict__ W,
                                                     float* __restrict__ h, int N) {
  __shared__ float lw[NF * NF];                 // 64 KB of the 320 KB WGP LDS
  int tid = threadIdx.x;
  // stage W into LDS with b128 loads/stores
  {
    const float4* Wv = (const float4*)W;
    float4* lv = (float4*)lw;
    for (int i = tid; i < NF * NF / 4; i += 256) lv[i] = Wv[i];
  }
  __syncthreads();

  int wave = tid >> 5;
  int lane = tid & 31;
  int lo16 = lane & 15;
  int hi   = lane >> 4;                         // 0: lanes 0-15, 1: lanes 16-31
  int m0   = blockIdx.x * 128 + wave * 16;
  int mrow = m0 + lo16;
  int mld  = (mrow < N) ? mrow : (N - 1);       // clamp OOB reads; stores guarded

  v8f acc[8] = {};
  const float* xr = x + (size_t)mld * NF;

  for (int k0 = 0; k0 < NF; k0 += 4) {
    // A fragment (16x4 f32): lane holds row M=lane%16; VGPR v <-> K = k0+v+2*hi
    int ka = k0 + 2 * hi;
    v2f a;
    a.x = xr[ka];
    a.y = xr[ka + 1];
#pragma unroll
    for (int nt = 0; nt < 8; ++nt) {
      // B fragment (4x16 f32): lane holds col N=nt*16+lane%16; VGPR v <-> K = k0+v+2*hi
      int nn = nt * 16 + lo16;
      v2f b;
      b.x = lw[ka * NF + nn];
      b.y = lw[(ka + 1) * NF + nn];
      acc[nt] = __builtin_amdgcn_wmma_f32_16x16x4_f32(
          /*neg_a=*/false, a, /*neg_b=*/false, b,
          /*c_mod=*/(short)0, acc[nt], /*reuse_a=*/false, /*reuse_b=*/false);
    }
  }

  // C/D layout: VGPR v <-> row M = v + 8*hi, col = lane%16
  if (m0 + 16 <= N) {
    // wave-uniform fast path: no per-element guards, stores coalesce in 64B runs
#pragma unroll
    for (int nt = 0; nt < 8; ++nt) {
#pragma unroll
      for (int v = 0; v < 8; ++v) {
        int m = m0 + v + 8 * hi;
        h[(size_t)m * NF + nt * 16 + lo16] = acc[nt][v];
      }
    }
  } else {
    // ragged last block only
#pragma unroll
    for (int nt = 0; nt < 8; ++nt) {
#pragma unroll
      for (int v = 0; v < 8; ++v) {
        int m = m0 + v + 8 * hi;
        if (m < N) h[(size_t)m * NF + nt * 16 + lo16] = acc[nt][v];
      }
    }
  }
}

// ---------------- edge scatter: dst[col] += src[row] * norm ---------------
// One wave per edge; 4 floats per lane (128 = 32*4).
__global__ void scatter_kernel(const float* __restrict__ src, float* __restrict__ dst,
                               const int* __restrict__ row, const int* __restrict__ col,
                               const float* __restrict__ norm, int E, int Etot) {
  int gw = (blockIdx.x * blockDim.x + threadIdx.x) >> 5;
  int lane = threadIdx.x & 31;
  if (gw >= Etot) return;
  int r = (gw < E) ? row[gw] : (gw - E);
  int c = (gw < E) ? col[gw] : (gw - E);
  float nrm = norm[gw];
  const float4* s = (const float4*)(src + (size_t)r * NF);
  float4 v = s[lane];                            // global_load_b128
  float* d = dst + (size_t)c * NF + lane * 4;
  atomAddF(d + 0, v.x * nrm);
  atomAddF(d + 1, v.y * nrm);
  atomAddF(d + 2, v.z * nrm);
  atomAddF(d + 3, v.w * nrm);
}

// ---------------- bias + relu (in place) ---------------------------------
__global__ void relu_bias_kernel(float* __restrict__ h, const float* __restrict__ b,
                                 long long total) {
  long long i = (long long)blockIdx.x * blockDim.x + threadIdx.x;
  if (i < total) {
    float v = h[i] + b[(int)(i & (NF - 1))];
    h[i] = fmaxf(v, 0.0f);
  }
}

// ---------------- pool: pooled[batch[n]] += agg2[n]; cnt[g] += 1 ---------
__global__ void pool_kernel(const float* __restrict__ agg2, const int* __restrict__ batch,
                            float* __restrict__ pooled, float* __restrict__ cnt, int N) {
  int gw = (blockIdx.x * blockDim.x + threadIdx.x) >> 5;
  int lane = threadIdx.x & 31;
  if (gw >= N) return;
  int g = batch[gw];
  const float4* s = (const float4*)(agg2 + (size_t)gw * NF);
  float4 v = s[lane];
  float* p = pooled + g * NF + lane * 4;
  atomAddF(p + 0, v.x);
  atomAddF(p + 1, v.y);
  atomAddF(p + 2, v.z);
  atomAddF(p + 3, v.w);
  if (lane == 0) atomAddF(&cnt[g], 1.0f);
}

// ---------------- final: [16,128]@[128,2] + b2, /cnt, log_softmax --------
__global__ void final_kernel(const float* __restrict__ pooled, const float* __restrict__ cnt,
                             const float* __restrict__ W2, const float* __restrict__ b2,
                             float* __restrict__ out) {
  int t = threadIdx.x;          // 32 threads: (g, c) = (t/2, t%2)
  int g = t >> 1, c = t & 1;
  float s = 0.0f;
  for (int k = 0; k < NF; ++k) s += pooled[g * NF + k] * W2[k * NC + c];
  float n  = fmaxf(cnt[g], 1.0f);
  float z  = (s + cnt[g] * b2[c]) / n;
  float zo = __shfl_xor(z, 1);                   // partner class logit (wave32)
  float m  = fmaxf(z, zo);
  float lse = m + logf(__expf(z - m) + __expf(zo - m));
  out[g * NC + c] = z - lse;
}

// ---------------------------------------------------------------------------
extern "C" void kernel_launch(void* const* d_in, const int* in_sizes, int n_in,
                              void* d_out, int out_size, void* d_ws, size_t ws_size,
                              hipStream_t stream) {
  const float* x     = (const float*)d_in[0];
  const int*   edge  = (const int*)d_in[1];   // [2, E] (JAX x64-off -> int32)
  const int*   batch = (const int*)d_in[2];
  const float* W1    = (const float*)d_in[3];
  const float* b1    = (const float*)d_in[4];
  const float* W2    = (const float*)d_in[5];
  const float* b2    = (const float*)d_in[6];
  float*       out   = (float*)d_out;

  int N    = in_sizes[0] / NF;
  int E    = in_sizes[1] / 2;
  int Etot = E + N;
  const int* row = edge;
  const int* col = edge + E;

  // workspace layout (floats). h1 buffer is reused as agg2 after scatter-1.
  float* ws     = (float*)d_ws;
  float* deg    = ws;                                  // N  (becomes dinv)
  float* norm   = deg + N;                             // Etot
  float* h1     = norm + Etot;                         // N*128  (later: agg2)
  float* agg1   = h1 + (size_t)N * NF;                 // N*128
  float* pooled = agg1 + (size_t)N * NF;               // 16*128
  float* cnt    = pooled + NG * NF;                    // 16
  float* agg2   = h1;

  hipMemsetAsync(deg, 0, (size_t)N * sizeof(float), stream);
  hipMemsetAsync(agg1, 0, (size_t)N * NF * sizeof(float), stream);
  hipMemsetAsync(pooled, 0, (NG * NF + NG) * sizeof(float), stream);

  deg_kernel <<<(E + 255) / 256, 256, 0, stream>>>(col, deg, E);
  dinv_kernel<<<(N + 255) / 256, 256, 0, stream>>>(deg, N);
  norm_kernel<<<(Etot + 255) / 256, 256, 0, stream>>>(row, col, deg, norm, E, Etot);

  gemm_wmma_f32<<<(N + 127) / 128, 256, 0, stream>>>(x, W1, h1, N);

  int scatter_blocks = (int)(((long long)Etot * 32 + 255) / 256);
  scatter_kernel<<<scatter_blocks, 256, 0, stream>>>(h1, agg1, row, col, norm, E, Etot);

  long long tot = (long long)N * NF;
  relu_bias_kernel<<<(int)((tot + 255) / 256), 256, 0, stream>>>(agg1, b1, tot);

  hipMemsetAsync(agg2, 0, (size_t)N * NF * sizeof(float), stream);  // h1 is dead now
  scatter_kernel<<<scatter_blocks, 256, 0, stream>>>(agg1, agg2, row, col, norm, E, Etot);

  int pool_blocks = (int)(((long long)N * 32 + 255) / 256);
  pool_kernel<<<pool_blocks, 256, 0, stream>>>(agg2, batch, pooled, cnt, N);

  final_kernel<<<1, 32, 0, stream>>>(pooled, cnt, W2, b2, out);
}